// pressureModule_47021301957216
// MI455X (gfx1250) — compile-verified
//
#include <hip/hip_runtime.h>
#include <stdint.h>

#define WAVES_PER_BLOCK 8

// ---------------------------------------------------------------------------
// Pass 1: build CSR row offsets rs[0..N] from the sorted edge-target array i.
// rs[n] = first edge whose target >= n;  rs[N] = E.
// Sorted i => each thread fills only the (usually empty) gap behind it.
// ---------------------------------------------------------------------------
__global__ __launch_bounds__(256) void build_row_offsets(
    const int* __restrict__ ei, uint32_t* __restrict__ rs, int E, int N) {
  int e = blockIdx.x * blockDim.x + threadIdx.x;
  if (e >= E) return;
  int cur  = ei[e];
  int prev = (e == 0) ? -1 : ei[e - 1];
  for (int n = prev + 1; n <= cur; ++n) rs[n] = (uint32_t)e;
  if (e == E - 1) {
    for (int n = cur + 1; n <= N; ++n) rs[n] = (uint32_t)E;
  }
}

// ---------------------------------------------------------------------------
// Pass 2: one wave32 per particle. Double-buffered async global->LDS staging
// of the particle's edge slab (j, q, dir), Wendland-C2 pressure gradient,
// deterministic butterfly wave reduction, lane 0 writes -acc/rhoi.
// ---------------------------------------------------------------------------
__global__ __launch_bounds__(256) void sph_pressure_gather(
    const int* __restrict__ ej, const float* __restrict__ Vj,
    const float2* __restrict__ dist, const float* __restrict__ rq,
    const float* __restrict__ supportPtr, const float* __restrict__ rhoi,
    const float* __restrict__ pres_i, const float* __restrict__ pres_j,
    const uint32_t* __restrict__ rs, float2* __restrict__ out, int N) {
  __shared__ int    sJ[2][WAVES_PER_BLOCK][32];
  __shared__ float  sQ[2][WAVES_PER_BLOCK][32];
  __shared__ float2 sD[2][WAVES_PER_BLOCK][32];

  const int w    = (int)(threadIdx.x >> 5);
  const int lane = (int)(threadIdx.x & 31);
  const int n    = (int)blockIdx.x * WAVES_PER_BLOCK + w;
  if (n >= N) return;  // uniform per wave

  const int start = (int)rs[n];
  const int end   = (int)rs[n + 1];
  const int deg   = end - start;

  float sx = 0.0f, sy = 0.0f;

  if (deg > 0) {
    const float h    = supportPtr[0];
    // dW/dq = -20 q (1-q)^3 * (7/pi) / h^3  => magC * q * (1-q)^3
    const float magC = -140.0f / 3.14159265358979323846f / (h * h * h);
    const float pin  = pres_i[n];
    const int   nch  = (deg + 31) >> 5;

    // Issue one chunk's async global->LDS loads (ASYNCcnt += 3).
    auto issue = [&](int c, int b) {
      int e  = start + (c << 5) + lane;
      int ec = (e < end) ? e : (end - 1);            // clamp: keep all lanes active
      uint32_t aJ = (uint32_t)(uintptr_t)&sJ[b][w][lane];
      uint32_t aQ = (uint32_t)(uintptr_t)&sQ[b][w][lane];
      uint32_t aD = (uint32_t)(uintptr_t)&sD[b][w][lane];
      uint32_t o4 = (uint32_t)ec * 4u;
      uint32_t o8 = (uint32_t)ec * 8u;
      asm volatile("global_load_async_to_lds_b32 %0, %1, %2"
                   :: "v"(aJ), "v"(o4), "s"(ej) : "memory");
      asm volatile("global_load_async_to_lds_b32 %0, %1, %2"
                   :: "v"(aQ), "v"(o4), "s"(rq) : "memory");
      asm volatile("global_load_async_to_lds_b64 %0, %1, %2"
                   :: "v"(aD), "v"(o8), "s"(dist) : "memory");
    };

    issue(0, 0);
    int buf = 0;
    for (int c = 0; c < nch; ++c) {
      if (c + 1 < nch) {
        issue(c + 1, buf ^ 1);
        // ASYNCcnt retires in order: <=3 outstanding => chunk c is in LDS.
        asm volatile("s_wait_asynccnt 3" ::: "memory");
      } else {
        asm volatile("s_wait_asynccnt 0" ::: "memory");
      }
      int    jj = sJ[buf][w][lane];
      float  q  = sQ[buf][w][lane];
      float2 d  = sD[buf][w][lane];
      int    e  = start + (c << 5) + lane;
      if (e < end) {
        float om  = 1.0f - q;
        float mag = magC * q * om * om * om;
        float wgt = (pin + pres_j[jj]) * Vj[jj] * mag;  // L2-resident gathers
        sx += wgt * d.x;
        sy += wgt * d.y;
      }
      buf ^= 1;
    }
  }

  // Deterministic wave32 butterfly reduction.
  for (int off = 16; off > 0; off >>= 1) {
    sx += __shfl_xor(sx, off, 32);
    sy += __shfl_xor(sy, off, 32);
  }

  if (lane == 0) {
    float inv = -1.0f / rhoi[n];
    out[n] = make_float2(sx * inv, sy * inv);
  }
}

extern "C" void kernel_launch(void* const* d_in, const int* in_sizes, int n_in,
                              void* d_out, int out_size, void* d_ws, size_t ws_size,
                              hipStream_t stream) {
  // setup_inputs() order:
  // 0:i 1:j 2:ri 3:rj 4:Vi 5:Vj 6:distances 7:radialDistances 8:support
  // 9:numParticles 10:eps 11:rhoi 12:rhoj 13:pi 14:pj
  const int*    ei     = (const int*)d_in[0];
  const int*    ej     = (const int*)d_in[1];
  const float*  Vj     = (const float*)d_in[5];
  const float2* dist   = (const float2*)d_in[6];
  const float*  rq     = (const float*)d_in[7];
  const float*  sup    = (const float*)d_in[8];
  const float*  rhoi   = (const float*)d_in[11];
  const float*  pres_i = (const float*)d_in[13];
  const float*  pres_j = (const float*)d_in[14];

  const int E = in_sizes[0];
  const int N = in_sizes[4];

  uint32_t* rs = (uint32_t*)d_ws;  // (N+1) row offsets in scratch

  build_row_offsets<<<(E + 255) / 256, 256, 0, stream>>>(ei, rs, E, N);

  int blocks = (N + WAVES_PER_BLOCK - 1) / WAVES_PER_BLOCK;
  sph_pressure_gather<<<blocks, 256, 0, stream>>>(
      ej, Vj, dist, rq, sup, rhoi, pres_i, pres_j, rs, (float2*)d_out, N);
}